// MessagePassing_80487687127300
// MI455X (gfx1250) — compile-verified
//
#include <hip/hip_runtime.h>

#ifndef __has_builtin
#define __has_builtin(x) 0
#endif

// CDNA5 async global->LDS copy path (gfx1250): guarded because the two
// toolchains (ROCm 7.2 clang-22 vs upstream clang-23) may differ.
#if __has_builtin(__builtin_amdgcn_global_load_async_to_lds_b32) && \
    __has_builtin(__builtin_amdgcn_s_wait_asynccnt)
#define USE_ASYNC_LDS 1
#else
#define USE_ASYNC_LDS 0
#endif

#define FEAT_D 64   // feature dim (fixed by reference)
#define TILE   256  // edges staged per block
#define BLOCK  256  // 8 waves of 32

#if USE_ASYNC_LDS
typedef __attribute__((address_space(1))) int glb_int_t;
typedef __attribute__((address_space(3))) int lds_int_t;
// Flat LDS addresses carry the LDS byte offset in the low 32 bits (ISA 10.2),
// so generic->AS(3) via integer truncation is the standard cast.
__device__ __forceinline__ lds_int_t* to_lds(int* p) {
  return (lds_int_t*)(unsigned)(unsigned long long)p;
}
__device__ __forceinline__ glb_int_t* to_glb(const int* p) {
  return (glb_int_t*)(unsigned long long)p;
}
#endif

// One wave per edge: lanes 0..31 cover the 64 features as float2.
// Edge indices for a 256-edge tile are staged in LDS (async DMA when
// available), then broadcast per wave via readfirstlane so gather/scatter
// addresses use scalar-base + lane-offset addressing.
__global__ __launch_bounds__(BLOCK) void
edge_scatter_kernel(const float* __restrict__ x,
                    const int* __restrict__ srcs,
                    const int* __restrict__ dsts,
                    float* __restrict__ out, int E) {
  __shared__ int s_src[TILE];
  __shared__ int s_dst[TILE];

  const int t    = threadIdx.x;
  const int lane = t & 31;
  const int wave = t >> 5;
  const int base = blockIdx.x * TILE;
  if (base >= E) return;
  const int n = (E - base) < TILE ? (E - base) : TILE;

#if USE_ASYNC_LDS
  if (t < n) {
    __builtin_amdgcn_global_load_async_to_lds_b32(
        to_glb(srcs + base + t), to_lds(&s_src[t]), 0, 0);
    __builtin_amdgcn_global_load_async_to_lds_b32(
        to_glb(dsts + base + t), to_lds(&s_dst[t]), 0, 0);
  }
  __builtin_amdgcn_s_wait_asynccnt(0);
#else
  if (t < n) {
    s_src[t] = srcs[base + t];
    s_dst[t] = dsts[base + t];
  }
#endif
  __syncthreads();

  const int fo = lane << 1;  // float offset within the 64-wide row
  for (int j = wave; j < n; j += (BLOCK / 32)) {
    // Prefetch the next edge's source row toward the WGP.
    const int jn = j + (BLOCK / 32);
    if (jn < n) {
      const int psrc = __builtin_amdgcn_readfirstlane(s_src[jn]);
      __builtin_prefetch(x + (long)psrc * FEAT_D, 0, 0);
    }

    const int src = __builtin_amdgcn_readfirstlane(s_src[j]);
    const int dst = __builtin_amdgcn_readfirstlane(s_dst[j]);

    const float2 v = *(const float2*)(x + (long)src * FEAT_D + fo);
    float* o = out + (long)dst * FEAT_D + fo;
    // Non-returning device-scope f32 atomic adds (L2 atomic units).
    __hip_atomic_fetch_add(o,     v.x, __ATOMIC_RELAXED, __HIP_MEMORY_SCOPE_AGENT);
    __hip_atomic_fetch_add(o + 1, v.y, __ATOMIC_RELAXED, __HIP_MEMORY_SCOPE_AGENT);
  }
}

__global__ __launch_bounds__(BLOCK) void
zero_f4_kernel(float4* __restrict__ o, int n4) {
  int i = blockIdx.x * BLOCK + threadIdx.x;
  if (i < n4) o[i] = make_float4(0.f, 0.f, 0.f, 0.f);
}

extern "C" void kernel_launch(void* const* d_in, const int* in_sizes, int n_in,
                              void* d_out, int out_size, void* d_ws, size_t ws_size,
                              hipStream_t stream) {
  (void)n_in; (void)d_ws; (void)ws_size;

  const float* x  = (const float*)d_in[0];
  const int*   ei = (const int*)d_in[1];     // [2, E]: row0 = src, row1 = dst
  const int    E  = in_sizes[1] / 2;
  float*       out = (float*)d_out;

  // Zero the (poisoned) output, vectorized as float4.
  const int n4 = out_size / 4;
  zero_f4_kernel<<<(n4 + BLOCK - 1) / BLOCK, BLOCK, 0, stream>>>(
      (float4*)out, n4);

  // Scatter-add all edges: one 256-edge tile per block.
  const int nb = (E + TILE - 1) / TILE;
  edge_scatter_kernel<<<nb, BLOCK, 0, stream>>>(x, ei, ei + E, out, E);
}